// ProCoUNLoss_75737453298132
// MI455X (gfx1250) — compile-verified
//
#include <hip/hip_runtime.h>
#include <math.h>

#ifndef M_PI
#define M_PI 3.14159265358979323846
#endif

typedef __attribute__((ext_vector_type(2))) float v2f;
typedef __attribute__((ext_vector_type(8))) float v8f;

#define DD 128      // feature dim
#define NU 63       // Bessel order = D/2 - 1

// ---------------------------------------------------------------------------
// Prep: row-normalize Ave into Bn (C x 128), and compute fsq[n] = ||f_n||^2.
// One block of 128 threads per row; rows [0,C) are Ave rows, [C, C+N) features.
// ---------------------------------------------------------------------------
__global__ __launch_bounds__(128) void proco_prep(
    const float* __restrict__ Ave, const float* __restrict__ feat,
    float* __restrict__ Bn, float* __restrict__ fsq, int C, int N) {
  __shared__ float red[4];
  int row = blockIdx.x;
  int t = threadIdx.x;  // 0..127 == feature dim index
  const float* src = (row < C) ? (Ave + (size_t)row * DD)
                               : (feat + (size_t)(row - C) * DD);
  float v = src[t];
  float s = v * v;
  // wave32 butterfly reduce
  #pragma unroll
  for (int off = 16; off > 0; off >>= 1) s += __shfl_xor(s, off, 32);
  if ((t & 31) == 0) red[t >> 5] = s;
  __syncthreads();
  float tot = red[0] + red[1] + red[2] + red[3];
  if (row < C) {
    float nrm = fmaxf(sqrtf(tot), 1e-12f);   // clip(norm, 1e-12) as in reference
    Bn[(size_t)row * DD + t] = v / nrm;
  } else if (t == 0) {
    fsq[row - C] = tot;                       // TEMP == 1.0
  }
}

// ---------------------------------------------------------------------------
// Main: per wave, one 16x16 tile of dot = F x Bn^T via V_WMMA_F32_16X16X4_F32,
// then kappa_new = sqrt(k^2 + 2k*dot + |f|^2), plain fp64 Miller recurrence
// (no per-step logs: x >= ~99 so B_0 <= ~1e69, safely inside double range),
// asymptotic log i0e, final logits.
// Block = 256 threads = 8 waves covering 16 rows x 128 classes.
// ---------------------------------------------------------------------------
__global__ __launch_bounds__(256) void proco_main(
    const float* __restrict__ F, const float* __restrict__ Bn,
    const float* __restrict__ kappa, const float* __restrict__ logc,
    const float* __restrict__ fsq, float* __restrict__ out, int C) {
  const int lane = threadIdx.x & 31;
  const int wave = threadIdx.x >> 5;
  const int col0 = blockIdx.x * 128 + wave * 16;   // class tile base
  const int row0 = blockIdx.y * 16;                // feature-row tile base
  const int m = lane & 15;
  const int koff = (lane < 16) ? 0 : 2;            // ISA A/B 16x4 f32 K split

  // A operand: row (row0+m), K = k+koff, k+koff+1  (float2, 8B aligned)
  const int arow = row0 + m;
  int bcol = col0 + m;                              // B operand: class (col0+m)
  if (bcol > C - 1) bcol = C - 1;                   // clamp padded classes
  const float* Aptr = F + (size_t)arow * DD + koff;
  const float* Bptr = Bn + (size_t)bcol * DD + koff;

  v8f acc = {};
  #pragma unroll
  for (int k = 0; k < DD; k += 4) {
    v2f a = *(const v2f*)(Aptr + k);
    v2f b = *(const v2f*)(Bptr + k);
    // D = A(16x4,f32) x B(4x16,f32) + C(16x16,f32)
    acc = __builtin_amdgcn_wmma_f32_16x16x4_f32(
        /*neg_a=*/false, a, /*neg_b=*/false, b,
        /*c_mod=*/(short)0, acc, /*reuse_a=*/false, /*reuse_b=*/false);
  }

  // C/D layout: lanes 0-15 -> N=lane, M=j; lanes 16-31 -> N=lane-16, M=j+8.
  // => per lane: ONE class column, EIGHT feature rows.
  const int mbase = (lane < 16) ? 0 : 8;
  const int c = col0 + m;
  const int cc = (c < C) ? c : C - 1;
  const float kap = kappa[cc];
  const float lgc = logc[cc];

  double x[8], w[8], bi[8], bip[8], b63[8];
  #pragma unroll
  for (int j = 0; j < 8; ++j) {
    int r = row0 + mbase + j;
    double dk = (double)kap;
    double x2 = dk * dk + 2.0 * dk * (double)acc[j] + (double)fsq[r];
    x2 = (x2 > 1e-24) ? x2 : 1e-24;
    x[j] = sqrt(x2);
    w[j] = 2.0 / x[j];
    bi[j] = 1.0;   // B_{2*NU} = 1
    bip[j] = 0.0;  // B_{2*NU+1} = 0
  }

  // Miller backward recurrence: B_{i-1} = (2i/x) B_i + B_{i+1}
  for (int i = 2 * NU; i >= NU + 1; --i) {
    double di = (double)i;
    #pragma unroll
    for (int j = 0; j < 8; ++j) {
      double t = fma(w[j] * di, bi[j], bip[j]);
      bip[j] = bi[j];
      bi[j] = t;
    }
  }
  #pragma unroll
  for (int j = 0; j < 8; ++j) b63[j] = bi[j];  // B_NU
  for (int i = NU; i >= 1; --i) {
    double di = (double)i;
    #pragma unroll
    for (int j = 0; j < 8; ++j) {
      double t = fma(w[j] * di, bi[j], bip[j]);
      bip[j] = bi[j];
      bi[j] = t;
    }
  }
  // bi[j] now holds B_0

  if (c < C) {
    #pragma unroll
    for (int j = 0; j < 8; ++j) {
      int r = row0 + mbase + j;
      double xv = x[j];
      // log i0e(x), large-x asymptotic (x >= ~99 here; rel err ~1e-11):
      // i0e(x) ~ (2*pi*x)^-1/2 * [1 + t + 4.5 t^2 + 37.5 t^3 + 459.375 t^4], t=1/(8x)
      double t = 1.0 / (8.0 * xv);
      double u = t * (1.0 + t * (4.5 + t * (37.5 + t * 459.375)));
      double li0e = -0.5 * log(2.0 * M_PI * xv) + log1p(u);
      double lnu = li0e + log(b63[j] / bi[j]);   // + log(B_NU/B_0)
      float xf = (float)xv;
      out[(size_t)r * C + c] =
          (float)lnu + xf - 63.0f * logf(xf + 1e-20f) - lgc;
    }
  }
}

// ---------------------------------------------------------------------------
extern "C" void kernel_launch(void* const* d_in, const int* in_sizes, int n_in,
                              void* d_out, int out_size, void* d_ws, size_t ws_size,
                              hipStream_t stream) {
  const float* features = (const float*)d_in[0];
  // d_in[1] = labels (unused by the deterministic core, matches reference)
  const float* Ave   = (const float*)d_in[2];
  const float* kappa = (const float*)d_in[3];
  const float* logc  = (const float*)d_in[4];
  const int N = in_sizes[1];  // 256
  const int C = in_sizes[3];  // 1000

  float* Bn  = (float*)d_ws;                 // C x 128 normalized prototypes
  float* fsq = Bn + (size_t)C * DD;          // N feature squared norms

  proco_prep<<<C + N, 128, 0, stream>>>(Ave, features, Bn, fsq, C, N);

  dim3 grid((C + 127) / 128, (N + 15) / 16);
  proco_main<<<grid, 256, 0, stream>>>(features, Bn, kappa, logc, fsq,
                                       (float*)d_out, C);
}